// FlashDeformAttn3D_78262894068379
// MI455X (gfx1250) — compile-verified
//
#include <hip/hip_runtime.h>

#define DM     256          // d_model (also K of every GEMM)
#define NH     8
#define DH     32
#define NL     4
#define NP     4
#define NQ     8192
#define NB     2
#define LEN_IN 149760

typedef __attribute__((ext_vector_type(2))) float v2f;
typedef __attribute__((ext_vector_type(8))) float v8f;

// ---------------------------------------------------------------------------
// Generic C[M,Nn] = A[M,256] * B[256,Nn] + bias[Nn], fp32, via WMMA f32 16x16x4.
// Block = 128 threads = 4 waves; block tile = 64 rows x 16 cols; K chunked by 64
// through LDS. M % 64 == 0, Nn % 16 == 0 (true for all call sites).
// All index math is 32-bit (every buffer < 2 GB).
// ---------------------------------------------------------------------------
__global__ __launch_bounds__(128) void gemm256_wmma_f32(
    const float* __restrict__ A, const float* __restrict__ B,
    const float* __restrict__ bias, float* __restrict__ C, int Nn)
{
    __shared__ float As[64 * 64];   // 16 KB
    __shared__ float Bs[64 * 16];   //  4 KB

    const int t    = threadIdx.x;
    const int lane = t & 31;
    const int wv   = t >> 5;                  // wave 0..3 -> row sub-tile
    const int row0 = blockIdx.x * 64;
    const int col0 = blockIdx.y * 16;

    const int m     = lane & 15;              // A row within tile / B,C column
    const int khalf = (lane >> 4) << 1;       // 0 for lanes 0-15, 2 for 16-31

    // Hoisted 32-bit staging indices (invariant across K chunks).
    uint32_t aIdx[8], sA[8], bIdx[8], sB[8];
    #pragma unroll
    for (int it = 0; it < 8; ++it) {
        int j  = it * 128 + t;                // float4 index into 64x16 grid
        int r  = j >> 4;
        int c4 = j & 15;
        aIdx[it] = (uint32_t)(row0 + r) * DM + (uint32_t)(c4 << 2);
        sA[it]   = (uint32_t)(r * 64 + (c4 << 2));
        int i  = it * 128 + t;                // scalar index into 64x16 grid
        int rb = i >> 4;
        int cb = i & 15;
        bIdx[it] = (uint32_t)rb * (uint32_t)Nn + (uint32_t)(col0 + cb);
        sB[it]   = (uint32_t)i;
    }
    const float* aFragRow = &As[(wv * 16 + m) * 64 + khalf];

    v8f acc = {};

    for (int kc = 0; kc < 4; ++kc) {          // K chunks of 64
        const uint32_t kofsA = (uint32_t)(kc * 64);
        const uint32_t kofsB = (uint32_t)(kc * 64) * (uint32_t)Nn;

        // --- stage A: 64 rows x 64 cols, float4-coalesced ---
        #pragma unroll
        for (int it = 0; it < 8; ++it) {
            float4 v = *reinterpret_cast<const float4*>(A + aIdx[it] + kofsA);
            *reinterpret_cast<float4*>(&As[sA[it]]) = v;
        }
        // --- stage B: 64 rows x 16 cols ---
        #pragma unroll
        for (int it = 0; it < 8; ++it) {
            Bs[sB[it]] = B[bIdx[it] + kofsB];
        }
        __syncthreads();

        // --- preload all fragments, then 16 back-to-back WMMAs ---
        v2f afrag[16], bfrag[16];
        #pragma unroll
        for (int k4 = 0; k4 < 16; ++k4) {
            // A 16x4 fragment: v0=K{0|2}, v1=K{1|3} per ISA table (8B aligned)
            afrag[k4] = *reinterpret_cast<const v2f*>(aFragRow + k4 * 4);
            const int kb = k4 * 4 + khalf;
            bfrag[k4].x = Bs[kb * 16 + m];
            bfrag[k4].y = Bs[kb * 16 + 16 + m];
        }
        #pragma unroll
        for (int k4 = 0; k4 < 16; ++k4) {
            acc = __builtin_amdgcn_wmma_f32_16x16x4_f32(
                false, afrag[k4], false, bfrag[k4], (short)0, acc, false, false);
        }
        __syncthreads();
    }

    // --- writeback: VGPR r holds M=r (lanes 0-15) / M=8+r (lanes 16-31) ---
    const int   ncol = lane & 15;
    const float bb   = bias[col0 + ncol];
    #pragma unroll
    for (int r = 0; r < 8; ++r) {
        int mrow = (lane < 16) ? r : (8 + r);
        C[(uint32_t)(row0 + wv * 16 + mrow) * (uint32_t)Nn + (uint32_t)(col0 + ncol)]
            = acc[r] + bb;
    }
}

// ---------------------------------------------------------------------------
// Deformable 3D sampling: one wave32 per (n, q, head); lane = channel (DH=32).
// Softmax over the 16 (level,point) logits is computed lane-redundantly from
// broadcast loads. Each corner gather is a 128B-coalesced read.
// ---------------------------------------------------------------------------
__global__ __launch_bounds__(128) void deform_sample3d(
    const float* __restrict__ refp,   // [NB, NQ, NL, 3]
    const float* __restrict__ value,  // [NB, LEN_IN, 256]
    const float* __restrict__ offw,   // [NB, NQ, NH*NL*NP*3]
    const float* __restrict__ attnw,  // [NB, NQ, NH*NL*NP] (logits)
    float* __restrict__ out)          // [NB, NQ, 256]
{
    const int lane = threadIdx.x & 31;
    const int wid  = blockIdx.x * (blockDim.x >> 5) + (threadIdx.x >> 5);
    if (wid >= NB * NQ * NH) return;

    const int h = wid & (NH - 1);
    const int q = (wid >> 3) & (NQ - 1);
    const int n = wid >> 16;

    constexpr int LD[NL]     = {32, 16, 8, 4};
    constexpr int LH[NL]     = {64, 32, 16, 8};
    constexpr int LW[NL]     = {64, 32, 16, 8};
    constexpr int LSTART[NL] = {0, 131072, 147456, 149504};

    const uint32_t qrow = (uint32_t)(n * NQ + q);

    // softmax over 16 logits (identical across lanes; broadcast loads)
    const float* lg = attnw + qrow * (NH * NL * NP) + h * (NL * NP);
    float l16[16];
    float mx = -3.0e38f;
    #pragma unroll
    for (int i = 0; i < 16; ++i) { l16[i] = lg[i]; mx = fmaxf(mx, l16[i]); }
    float s = 0.0f;
    #pragma unroll
    for (int i = 0; i < 16; ++i) { l16[i] = __expf(l16[i] - mx); s += l16[i]; }
    const float inv = 1.0f / s;

    const float* ob = offw + qrow * (NH * NL * NP * 3) + h * (NL * NP * 3);
    const float* rp = refp + qrow * (NL * 3);
    const float* vb = value + (uint32_t)n * (LEN_IN * DM) + (h * DH + lane);

    float acc = 0.0f;
    #pragma unroll
    for (int l = 0; l < NL; ++l) {
        const int D = LD[l], H = LH[l], W = LW[l];
        const float rx = rp[l * 3 + 0] * (float)W;
        const float ry = rp[l * 3 + 1] * (float)H;
        const float rz = rp[l * 3 + 2] * (float)D;
        const float* vlev = vb + (uint32_t)LSTART[l] * DM;
        #pragma unroll
        for (int p = 0; p < NP; ++p) {
            const int   lp = l * NP + p;
            const float aw = l16[lp] * inv;
            // loc = rp + off/norm; x = loc.x*W - 0.5 == rp.x*W + off.x - 0.5
            const float x = rx + ob[lp * 3 + 0] - 0.5f;
            const float y = ry + ob[lp * 3 + 1] - 0.5f;
            const float z = rz + ob[lp * 3 + 2] - 0.5f;
            const float xf = floorf(x), yf = floorf(y), zf = floorf(z);
            const float fx = x - xf, fy = y - yf, fz = z - zf;
            const int ix = (int)xf, iy = (int)yf, iz = (int)zf;
            #pragma unroll
            for (int c = 0; c < 8; ++c) {
                const int dx = c & 1, dy = (c >> 1) & 1, dz = c >> 2;
                const int xi = ix + dx, yi = iy + dy, zi = iz + dz;
                if (xi < 0 || xi >= W || yi < 0 || yi >= H || zi < 0 || zi >= D)
                    continue;   // wave-uniform branch
                const float w = (dx ? fx : 1.0f - fx) *
                                (dy ? fy : 1.0f - fy) *
                                (dz ? fz : 1.0f - fz);
                const float g = vlev[(uint32_t)((zi * H + yi) * W + xi) * DM];
                acc = fmaf(aw * w, g, acc);
            }
        }
    }
    out[qrow * DM + (uint32_t)(h * DH + lane)] = acc;
}

// ---------------------------------------------------------------------------
extern "C" void kernel_launch(void* const* d_in, const int* in_sizes, int n_in,
                              void* d_out, int out_size, void* d_ws, size_t ws_size,
                              hipStream_t stream)
{
    (void)in_sizes; (void)n_in; (void)out_size; (void)ws_size;

    const float* query  = (const float*)d_in[0];
    const float* refp   = (const float*)d_in[1];
    const float* inflat = (const float*)d_in[2];
    // d_in[3]/d_in[4]: spatial shapes / level starts — compile-time constants here
    const float* Wv     = (const float*)d_in[5];
    const float* bv     = (const float*)d_in[6];
    const float* Woff   = (const float*)d_in[7];
    const float* boff   = (const float*)d_in[8];
    const float* Wattn  = (const float*)d_in[9];
    const float* battn  = (const float*)d_in[10];
    const float* Wout   = (const float*)d_in[11];
    const float* bout   = (const float*)d_in[12];
    float* out = (float*)d_out;

    // workspace layout (floats)
    float* ws    = (float*)d_ws;
    float* value = ws;                                        // NB*LEN_IN*256
    float* offb  = value + (size_t)NB * LEN_IN * DM;          // NB*NQ*384
    float* attnb = offb  + (size_t)NB * NQ * (NH * NL * NP * 3); // NB*NQ*128
    float* sout  = attnb + (size_t)NB * NQ * (NH * NL * NP);  // NB*NQ*256

    // 1) value = input_flatten @ Wv + bv        (299520 x 256 x 256)
    gemm256_wmma_f32<<<dim3((NB * LEN_IN) / 64, DM / 16), 128, 0, stream>>>(
        inflat, Wv, bv, value, DM);

    // 2) offsets = query @ Woff + boff          (16384 x 384 x 256)
    gemm256_wmma_f32<<<dim3((NB * NQ) / 64, 384 / 16), 128, 0, stream>>>(
        query, Woff, boff, offb, 384);

    // 3) attn logits = query @ Wattn + battn    (16384 x 128 x 256)
    gemm256_wmma_f32<<<dim3((NB * NQ) / 64, 128 / 16), 128, 0, stream>>>(
        query, Wattn, battn, attnb, 128);

    // 4) softmax + trilinear sampling + point/level accumulation
    deform_sample3d<<<(NB * NQ * NH) / 4, 128, 0, stream>>>(
        refp, value, offb, attnb, sout);

    // 5) out = sout @ Wout + bout               (16384 x 256 x 256)
    gemm256_wmma_f32<<<dim3((NB * NQ) / 64, DM / 16), 128, 0, stream>>>(
        sout, Wout, bout, out, DM);
}